// DGL_GAT_50697793962360
// MI455X (gfx1250) — compile-verified
//
#include <hip/hip_runtime.h>
#include <hip/hip_bf16.h>
#include <math.h>

typedef __attribute__((ext_vector_type(2))) float v2f;
typedef __attribute__((ext_vector_type(8))) float v8f;

#define NEG_SLOPE 0.2f

// ---------------------------------------------------------------------------
// GEMM: C(N,M) = A(N,K) * W(M,K)^T  using V_WMMA_F32_16X16X4_F32.
// One wave (32 threads) computes one 16x16 tile of C, accumulating over K in
// steps of 4. A-fragment: lanes 0-15 hold {K=k,k+1} of row (lane&15); lanes
// 16-31 hold {K=k+2,k+3}. B-fragment mirrors with columns. Both are
// contiguous float2 loads since A and W are row-major over K.
// ---------------------------------------------------------------------------
__global__ __launch_bounds__(32) void gat_gemm_wmma_f32(
    const float* __restrict__ A,   // (N, K) row-major
    const float* __restrict__ W,   // (M, K) row-major
    float* __restrict__ C,         // (N, M) row-major
    int N, int K, int M)
{
    const int rowTile = blockIdx.x * 16;
    const int colTile = blockIdx.y * 16;
    const int lane    = threadIdx.x;
    const int r       = lane & 15;
    const int koff    = (lane >> 4) * 2;   // 0 or 2

    int arow = rowTile + r;
    if (arow > N - 1) arow = N - 1;        // clamp (keeps EXEC all-1 for WMMA)
    const float* __restrict__ ap = A + (size_t)arow * K;
    const float* __restrict__ wp = W + (size_t)(colTile + r) * K;

    v8f acc = {};
    for (int k = 0; k < K; k += 4) {
        v2f a = *(const v2f*)(ap + k + koff);
        v2f b = *(const v2f*)(wp + k + koff);
        acc = __builtin_amdgcn_wmma_f32_16x16x4_f32(
            /*neg_a=*/false, a, /*neg_b=*/false, b,
            /*c_mod=*/(short)0, acc, /*reuse_a=*/false, /*reuse_b=*/false);
    }

    // C/D layout: VGPR j holds row j (lanes 0-15) or row j+8 (lanes 16-31),
    // column = lane & 15.
    const int rbase = rowTile + (lane >> 4) * 8;
    const int col   = colTile + (lane & 15);
#pragma unroll
    for (int j = 0; j < 8; ++j) {
        int rr = rbase + j;
        if (rr < N) C[(size_t)rr * M + col] = acc[j];
    }
}

// el[n] = feat[n,:]·al ; er[n] = feat[n,:]·ar
__global__ void gat_attn_proj(const float* __restrict__ feat,
                              const float* __restrict__ al,
                              const float* __restrict__ ar,
                              float* __restrict__ el,
                              float* __restrict__ er,
                              int N, int D)
{
    int n = blockIdx.x * blockDim.x + threadIdx.x;
    if (n >= N) return;
    const float* f = feat + (size_t)n * D;
    float sl = 0.f, sr = 0.f;
    for (int k = 0; k < D; ++k) {
        float v = f[k];
        sl = fmaf(v, al[k], sl);
        sr = fmaf(v, ar[k], sr);
    }
    el[n] = sl;
    er[n] = sr;
}

// out[n,k] = bias[k]; emax[n] = -inf; denom[n] = 0
__global__ void gat_init(float* __restrict__ out,
                         const float* __restrict__ bias,
                         float* __restrict__ emax,
                         float* __restrict__ denom,
                         int N, int D)
{
    int idx = blockIdx.x * blockDim.x + threadIdx.x;
    if (idx >= N * D) return;
    int n = idx / D;
    int k = idx - n * D;
    out[idx] = bias[k];
    if (k == 0) {
        emax[n]  = -INFINITY;
        denom[n] = 0.f;
    }
}

// e[i] = leaky_relu(el[src]+er[dst]); segment-max into emax[dst] (int trick)
__global__ void gat_edge_logits(const float* __restrict__ el,
                                const float* __restrict__ er,
                                const int* __restrict__ src,
                                const int* __restrict__ dst,
                                float* __restrict__ e,
                                float* __restrict__ emax,
                                int E)
{
    int i = blockIdx.x * blockDim.x + threadIdx.x;
    if (i >= E) return;
    int s = src[i], d = dst[i];
    float v = el[s] + er[d];
    v = v > 0.f ? v : NEG_SLOPE * v;
    e[i] = v;
    // float atomic max via signed-max (v>=0) / unsigned-min (v<0); init -inf.
    if (v >= 0.f)
        atomicMax((int*)emax + d, __float_as_int(v));
    else
        atomicMin((unsigned int*)emax + d, __float_as_uint(v));
}

// ee[i] = exp(e[i]-emax[dst]); denom[dst] += ee[i]
__global__ void gat_edge_exp(const float* __restrict__ e,
                             const int* __restrict__ dst,
                             const float* __restrict__ emax,
                             float* __restrict__ ee,
                             float* __restrict__ denom,
                             int E)
{
    int i = blockIdx.x * blockDim.x + threadIdx.x;
    if (i >= E) return;
    int d = dst[i];
    float x = expf(e[i] - emax[d]);
    ee[i] = x;
    atomicAdd(&denom[d], x);
}

// one wave per edge; lane k handles feature k:
// out[dst,k] += (ee/denom[dst]) * feat[src,k]
__global__ void gat_edge_scatter(const float* __restrict__ feat,
                                 const int* __restrict__ src,
                                 const int* __restrict__ dst,
                                 const float* __restrict__ ee,
                                 const float* __restrict__ denom,
                                 float* __restrict__ out,
                                 int E, int D)
{
    int gid = blockIdx.x * blockDim.x + threadIdx.x;
    int i = gid / D;
    int k = gid - i * D;
    if (i >= E) return;
    int s = src[i], d = dst[i];
    float alpha = ee[i] / denom[d];
    atomicAdd(&out[(size_t)d * D + k], alpha * feat[(size_t)s * D + k]);
}

// ---------------------------------------------------------------------------
// Host-side layer driver (all launches on `stream`; graph-capture safe).
// ---------------------------------------------------------------------------
static void gat_run_layer(const float* h, int Din,
                          const float* W, const float* al, const float* ar,
                          const float* bias,
                          const int* src, const int* dst,
                          int N, int E, int Dout,
                          float* feat, float* el, float* er,
                          float* e, float* ee, float* emax, float* denom,
                          float* out, hipStream_t stream)
{
    dim3 gg((N + 15) / 16, Dout / 16);
    gat_gemm_wmma_f32<<<gg, 32, 0, stream>>>(h, W, feat, N, Din, Dout);

    gat_attn_proj<<<(N + 255) / 256, 256, 0, stream>>>(feat, al, ar, el, er, N, Dout);

    int nd = N * Dout;
    gat_init<<<(nd + 255) / 256, 256, 0, stream>>>(out, bias, emax, denom, N, Dout);

    gat_edge_logits<<<(E + 255) / 256, 256, 0, stream>>>(el, er, src, dst, e, emax, E);
    gat_edge_exp<<<(E + 255) / 256, 256, 0, stream>>>(e, dst, emax, ee, denom, E);

    long long ed = (long long)E * Dout;
    gat_edge_scatter<<<(unsigned)((ed + 255) / 256), 256, 0, stream>>>(
        feat, src, dst, ee, denom, out, E, Dout);
}

extern "C" void kernel_launch(void* const* d_in, const int* in_sizes, int n_in,
                              void* d_out, int out_size, void* d_ws, size_t ws_size,
                              hipStream_t stream)
{
    const float* features = (const float*)d_in[0];
    const int*   src      = (const int*)d_in[1];
    const int*   dst      = (const int*)d_in[2];
    const float* W1  = (const float*)d_in[3];
    const float* al1 = (const float*)d_in[4];
    const float* ar1 = (const float*)d_in[5];
    const float* b1  = (const float*)d_in[6];
    const float* W2  = (const float*)d_in[7];
    const float* al2 = (const float*)d_in[8];
    const float* ar2 = (const float*)d_in[9];
    const float* b2  = (const float*)d_in[10];

    const int H    = in_sizes[4];            // n_hidden (32)
    const int Din  = in_sizes[3] / H;        // 128
    const int N    = in_sizes[0] / Din;      // 100000
    const int E    = in_sizes[1];            // 1600000
    const int Dout = in_sizes[8];            // 32

    // Workspace carve-out (floats), 16B-aligned chunks.
    float* ws = (float*)d_ws;
    size_t off = 0;
    auto carve = [&](size_t n) {
        float* p = ws + off;
        off += (n + 3) & ~(size_t)3;
        return p;
    };
    const int Dmax = H > Dout ? H : Dout;
    float* feat  = carve((size_t)N * Dmax);  // GEMM output (per layer)
    float* el    = carve((size_t)N);
    float* er    = carve((size_t)N);
    float* emax  = carve((size_t)N);
    float* denom = carve((size_t)N);
    float* e     = carve((size_t)E);
    float* ee    = carve((size_t)E);
    float* h1    = carve((size_t)N * H);     // layer-1 GAT output

    // Layer 1: features(N,128) -> h1(N,H)
    gat_run_layer(features, Din, W1, al1, ar1, b1, src, dst, N, E, H,
                  feat, el, er, e, ee, emax, denom, h1, stream);

    // Layer 2: h1(N,H) -> d_out(N,Dout)
    gat_run_layer(h1, H, W2, al2, ar2, b2, src, dst, N, E, Dout,
                  feat, el, er, e, ee, emax, denom, (float*)d_out, stream);
}